// AuxEntTypeTower_75222057222947
// MI455X (gfx1250) — compile-verified
//
#include <hip/hip_runtime.h>
#include <math.h>

typedef float v2f __attribute__((ext_vector_type(2)));
typedef float v8f __attribute__((ext_vector_type(8)));

constexpr int Bn   = 4;
constexpr int Ln   = 1024;
constexpr int Hn   = 512;     // K
constexpr int Tn   = 8192;    // N (power of 2: 2^13)
constexpr int Mtot = Bn * Ln; // 4096 (M)

constexpr int NBLK    = 2048; // partial-reduction blocks
constexpr int THREADS = 256;

// ---------------------------------------------------------------------------
// Kernel 1: logits = feature @ head_w^T + head_b via V_WMMA_F32_16X16X4_F32
// Block = 256 threads (8 wave32). Block tile: 128M x 128N.
// Wave (w&3) selects 32-row M slot, (w>>2) selects 64-col N slot.
// A fragment (16x4 f32): lanes 0-15 rows M, VGPR0/1 = K{0,1} (lanes<16) or
// K{2,3} (lanes>=16)  -> contiguous float2 load at feature[m][k + 2*(lane>>4)].
// B fragment (4x16 f32) mirrors A with N across lanes; head_w is [T,H]
// row-major so B is also a contiguous float2 load at head_w[n][k + 2*(lane>>4)].
// ---------------------------------------------------------------------------
__global__ __launch_bounds__(256)
void gemm_logits_wmma(const float* __restrict__ feature,
                      const float* __restrict__ head_w,
                      const float* __restrict__ head_b,
                      float* __restrict__ logits) {
  const int tid   = threadIdx.x;
  const int wave  = tid >> 5;
  const int lane  = tid & 31;
  const int l16   = lane & 15;
  const int lhalf = lane >> 4;      // 0: K{0,1}, 1: K{2,3}
  const int kl    = lhalf * 2;

  const int m0 = blockIdx.y * 128 + (wave & 3) * 32;
  const int n0 = blockIdx.x * 128 + (wave >> 2) * 64;

  const float* A0 = feature + (size_t)(m0 + l16) * Hn + kl;
  const float* A1 = A0 + (size_t)16 * Hn;
  const float* Bp = head_w + (size_t)(n0 + l16) * Hn + kl;

  const v8f zero = {};
  v8f acc[2][4];
#pragma unroll
  for (int i = 0; i < 2; ++i)
#pragma unroll
    for (int j = 0; j < 4; ++j) acc[i][j] = zero;

#pragma unroll 4
  for (int k = 0; k < Hn; k += 4) {
    v2f a0 = *(const v2f*)(A0 + k);
    v2f a1 = *(const v2f*)(A1 + k);
    v2f b[4];
#pragma unroll
    for (int j = 0; j < 4; ++j)
      b[j] = *(const v2f*)(Bp + (size_t)j * 16 * Hn + k);
#pragma unroll
    for (int j = 0; j < 4; ++j) {
      acc[0][j] = __builtin_amdgcn_wmma_f32_16x16x4_f32(
          false, a0, false, b[j], (short)0, acc[0][j], false, false);
      acc[1][j] = __builtin_amdgcn_wmma_f32_16x16x4_f32(
          false, a1, false, b[j], (short)0, acc[1][j], false, false);
    }
  }

  // D layout (16x16 f32): VGPR r -> row m0 + 8*lhalf + r, col n0 + l16.
#pragma unroll
  for (int j = 0; j < 4; ++j) {
    const float bias = head_b[n0 + j * 16 + l16];
#pragma unroll
    for (int i = 0; i < 2; ++i) {
      float* C = logits + (size_t)(m0 + i * 16 + lhalf * 8) * Tn + n0 + j * 16 + l16;
#pragma unroll
      for (int r = 0; r < 8; ++r)
        C[(size_t)r * Tn] = acc[i][j][r] + bias;
    }
  }
}

// ---------------------------------------------------------------------------
// Kernel 2: masked BCE-with-logits against soft target, block partial sums.
// prev[l] = sigmoid(logits[l-1]) for l>0; for l==0 prev == target (same index).
// ---------------------------------------------------------------------------
__global__ __launch_bounds__(THREADS)
void loss_partial(const float* __restrict__ logits,
                  const float* __restrict__ target,
                  const unsigned char* __restrict__ mask,
                  const float* __restrict__ alpha_p,
                  float* __restrict__ partial) {
  const float alpha = *alpha_p;
  const float oma   = 1.0f - alpha;
  const long long total  = (long long)Mtot * Tn;
  const long long stride = (long long)NBLK * THREADS;

  float sum = 0.0f;
  for (long long i = (long long)blockIdx.x * THREADS + threadIdx.x;
       i < total; i += stride) {
    const long long bl = i >> 13;            // / Tn
    const int l = (int)(bl & (Ln - 1));
    if (mask[bl]) {
      const float x   = logits[i];
      const float tgt = target[i];
      float prev;
      if (l == 0) {
        prev = tgt;                          // target[b, 0, t]
      } else {
        const float xp = logits[i - Tn];
        prev = 1.0f / (1.0f + expf(-xp));
      }
      const float st = oma * tgt + alpha * prev;
      const float sp = fmaxf(x, 0.0f) + log1pf(expf(-fabsf(x)));  // softplus
      sum += sp - x * st;
    }
  }

  __shared__ float red[THREADS];
  red[threadIdx.x] = sum;
  __syncthreads();
#pragma unroll
  for (int s = THREADS / 2; s > 0; s >>= 1) {
    if (threadIdx.x < s) red[threadIdx.x] += red[threadIdx.x + s];
    __syncthreads();
  }
  if (threadIdx.x == 0) partial[blockIdx.x] = red[0];
}

// ---------------------------------------------------------------------------
// Kernel 3: reduce partials, divide by mask count, write scalar loss.
// ---------------------------------------------------------------------------
__global__ __launch_bounds__(256)
void loss_final(const float* __restrict__ partial,
                const unsigned char* __restrict__ mask,
                float* __restrict__ loss_out) {
  __shared__ float reds[256];
  __shared__ float redm[256];
  float s  = 0.0f;
  for (int i = threadIdx.x; i < NBLK; i += 256) s += partial[i];
  float mc = 0.0f;
  for (int i = threadIdx.x; i < Mtot; i += 256) mc += mask[i] ? 1.0f : 0.0f;
  reds[threadIdx.x] = s;
  redm[threadIdx.x] = mc;
  __syncthreads();
#pragma unroll
  for (int st = 128; st > 0; st >>= 1) {
    if (threadIdx.x < st) {
      reds[threadIdx.x] += reds[threadIdx.x + st];
      redm[threadIdx.x] += redm[threadIdx.x + st];
    }
    __syncthreads();
  }
  if (threadIdx.x == 0) loss_out[0] = reds[0] / fmaxf(redm[0], 1.0f);
}

// ---------------------------------------------------------------------------
extern "C" void kernel_launch(void* const* d_in, const int* in_sizes, int n_in,
                              void* d_out, int out_size, void* d_ws, size_t ws_size,
                              hipStream_t stream) {
  const float*         feature = (const float*)d_in[0];
  const float*         target  = (const float*)d_in[1];
  const unsigned char* mask    = (const unsigned char*)d_in[2];  // jnp.bool_ (1B)
  const float*         head_w  = (const float*)d_in[3];
  const float*         head_b  = (const float*)d_in[4];
  const float*         alpha   = (const float*)d_in[5];

  float* logits  = (float*)d_out;                      // [B,L,T] = 33554432 floats
  float* loss    = logits + (size_t)Mtot * Tn;         // final scalar
  float* partial = (float*)d_ws;                       // NBLK floats scratch

  dim3 grid(Tn / 128, Mtot / 128);                     // 64 x 32 blocks
  gemm_logits_wmma<<<grid, 256, 0, stream>>>(feature, head_w, head_b, logits);
  loss_partial<<<NBLK, THREADS, 0, stream>>>(logits, target, mask, alpha, partial);
  loss_final<<<1, 256, 0, stream>>>(partial, mask, loss);
}